// Router_13288628814473
// MI455X (gfx1250) — compile-verified
//
#include <hip/hip_runtime.h>
#include <hip/hip_bf16.h>

typedef __attribute__((ext_vector_type(2))) float v2f;
typedef __attribute__((ext_vector_type(8))) float v8f;

#define D_MODEL     1024
#define N_EXP       64
#define TOK_PER_BLK 128
#define KC          32
#define LA_STRIDE   36    // words/row: 144B -> 16B aligned, A-frag banks are a permutation
#define LB_PSTRIDE  160   // words per K-pair row: 160 mod 64 = 32 -> half-waves on disjoint banks
#define LS_STRIDE   66    // logits rows: half-wave spill writes on disjoint banks

#define LA_WORDS (TOK_PER_BLK * LA_STRIDE)   // 4608
#define LB_WORDS ((KC / 2) * LB_PSTRIDE)     // 2560

__global__ __launch_bounds__(256) void router_wmma_kernel(
    const float* __restrict__ x, const float* __restrict__ Wg,
    float* __restrict__ outCombine, int* __restrict__ outIdx,
    float* __restrict__ outProbs)
{
    // Stage buffers (double-buffered) overlaid with the logits scratch:
    // stage = 2*(18432 + 10240) = 57344 B; logits = 33792 B -> union keeps us < 64 KB.
    __shared__ union {
        struct { float A[2][LA_WORDS]; float B[2][LB_WORDS]; } st;
        float S[TOK_PER_BLK * LS_STRIDE];
    } u;
    __shared__ float sMax[TOK_PER_BLK];
    __shared__ float sRcp[TOK_PER_BLK];

    const int tid    = threadIdx.x;
    const int lane   = tid & 31;
    const int wave   = tid >> 5;
    const int blkTok = blockIdx.x * TOK_PER_BLK;

    const int m    = lane & 15;   // A row / B col within 16-tile
    const int half = lane >> 4;   // 0: K=k,k+1   1: K=k+2,k+3
    const int koff = half * 2;

    v8f acc0 = {}, acc1 = {}, acc2 = {}, acc3 = {};

    // --- staging helpers -------------------------------------------------
    const int pB  = tid >> 4;          // K-pair row this thread stages (0..15)
    const int n0  = (tid & 15) * 4;    // expert column base (0,4,...,60)

    auto loadStage = [&](int kc, float4* ra, float4& rb0, float4& rb1) {
        #pragma unroll
        for (int i = 0; i < 4; ++i) {
            int flat = i * 1024 + tid * 4;
            int r = flat >> 5, c = flat & 31;
            ra[i] = *(const float4*)&x[(size_t)(blkTok + r) * D_MODEL + kc + c];
        }
        rb0 = *(const float4*)&Wg[(size_t)(kc + 2 * pB + 0) * N_EXP + n0];
        rb1 = *(const float4*)&Wg[(size_t)(kc + 2 * pB + 1) * N_EXP + n0];
    };

    auto storeStage = [&](int buf, const float4* ra, float4 rb0, float4 rb1) {
        #pragma unroll
        for (int i = 0; i < 4; ++i) {
            int flat = i * 1024 + tid * 4;
            int r = flat >> 5, c = flat & 31;
            *(float4*)&u.st.A[buf][r * LA_STRIDE + c] = ra[i];
        }
        // K-pair interleave: value (k, n) -> B[k/2][n*2 + (k&1)]
        float4 w0 = { rb0.x, rb1.x, rb0.y, rb1.y };
        float4 w1 = { rb0.z, rb1.z, rb0.w, rb1.w };
        *(float4*)&u.st.B[buf][pB * LB_PSTRIDE + n0 * 2 + 0] = w0;
        *(float4*)&u.st.B[buf][pB * LB_PSTRIDE + n0 * 2 + 4] = w1;
    };

    auto compute = [&](int buf) {
        const float* aRow  = &u.st.A[buf][(wave * 16 + m) * LA_STRIDE];
        const float* bHalf = &u.st.B[buf][half * LB_PSTRIDE + m * 2];
        #pragma unroll
        for (int k4 = 0; k4 < KC; k4 += 4) {
            v2f a = *(const v2f*)&aRow[k4 + koff];                 // ds_load_b64
            const float* bp = bHalf + (k4 >> 1) * LB_PSTRIDE;
            v2f b0 = *(const v2f*)(bp +  0);                       // ds_load_b64, pair-adjacent
            v2f b1 = *(const v2f*)(bp + 32);
            v2f b2 = *(const v2f*)(bp + 64);
            v2f b3 = *(const v2f*)(bp + 96);
            acc0 = __builtin_amdgcn_wmma_f32_16x16x4_f32(false, a, false, b0, (short)0, acc0, false, false);
            acc1 = __builtin_amdgcn_wmma_f32_16x16x4_f32(false, a, false, b1, (short)0, acc1, false, false);
            acc2 = __builtin_amdgcn_wmma_f32_16x16x4_f32(false, a, false, b2, (short)0, acc2, false, false);
            acc3 = __builtin_amdgcn_wmma_f32_16x16x4_f32(false, a, false, b3, (short)0, acc3, false, false);
        }
    };

    // --- prologue + double-buffered main loop ----------------------------
    float4 ra[4], rb0, rb1;
    loadStage(0, ra, rb0, rb1);
    storeStage(0, ra, rb0, rb1);
    __syncthreads();

    #pragma unroll 1
    for (int kc = 0; kc < D_MODEL; kc += KC) {
        const int cur = (kc >> 5) & 1;
        const bool more = (kc + KC) < D_MODEL;
        if (more) loadStage(kc + KC, ra, rb0, rb1);   // global loads in flight over compute
        compute(cur);
        if (more) storeStage(cur ^ 1, ra, rb0, rb1);
        __syncthreads();
    }

    // --- spill logits (D layout: VGPR j -> row j / j+8) -------------------
    {
        const int trow = wave * 16 + half * 8;
        #pragma unroll
        for (int j = 0; j < 8; ++j) {
            u.S[(trow + j) * LS_STRIDE +  0 + m] = acc0[j];
            u.S[(trow + j) * LS_STRIDE + 16 + m] = acc1[j];
            u.S[(trow + j) * LS_STRIDE + 32 + m] = acc2[j];
            u.S[(trow + j) * LS_STRIDE + 48 + m] = acc3[j];
        }
    }
    __syncthreads();

    // --- per-token softmax stats + top-2 + combine ------------------------
    if (tid < TOK_PER_BLK) {
        const float* row = &u.S[tid * LS_STRIDE];
        float mx = row[0];
        #pragma unroll
        for (int e = 1; e < N_EXP; ++e) mx = fmaxf(mx, row[e]);
        float l1 = -3.402823466e38f, l2 = -3.402823466e38f;
        int i1 = 0, i2 = 0;
        float s = 0.0f;
        #pragma unroll
        for (int e = 0; e < N_EXP; ++e) {
            float v = row[e];
            s += __expf(v - mx);
            if (v > l1)      { l2 = l1; i2 = i1; l1 = v; i1 = e; }
            else if (v > l2) { l2 = v;  i2 = e; }
        }
        sMax[tid] = mx;
        sRcp[tid] = 1.0f / s;
        float c0 = 1.0f / (1.0f + __expf(l2 - l1));   // p1/(p1+p2), numerically stable
        size_t g = (size_t)blkTok + tid;
        outCombine[g * 2 + 0] = c0;
        outCombine[g * 2 + 1] = 1.0f - c0;
        outIdx[g * 2 + 0] = i1;
        outIdx[g * 2 + 1] = i2;
    }
    __syncthreads();

    // --- probs matrix: contiguous 32 KB per block, coalesced float4 stores
    {
        const size_t base = (size_t)blkTok * N_EXP;
        #pragma unroll
        for (int i = 0; i < 8; ++i) {
            int flat = i * 1024 + tid * 4;
            int t = flat >> 6, e = flat & 63;
            float mx = sMax[t], r = sRcp[t];
            const float* row = &u.S[t * LS_STRIDE + e];
            float4 v;
            v.x = __expf(row[0] - mx) * r;
            v.y = __expf(row[1] - mx) * r;
            v.z = __expf(row[2] - mx) * r;
            v.w = __expf(row[3] - mx) * r;
            *(float4*)&outProbs[base + flat] = v;
        }
    }
}

extern "C" void kernel_launch(void* const* d_in, const int* in_sizes, int n_in,
                              void* d_out, int out_size, void* d_ws, size_t ws_size,
                              hipStream_t stream) {
    const float* x  = (const float*)d_in[0];
    const float* Wg = (const float*)d_in[1];
    const int T = in_sizes[0] / D_MODEL;   // 65536

    // d_out layout (return order): combine [T,2] f32 | topk_idx [T,2] i32 | probs [T,64] f32
    float* outCombine = (float*)d_out;
    int*   outIdx     = (int*)d_out + (size_t)2 * T;
    float* outProbs   = (float*)d_out + (size_t)4 * T;

    const int grid = T / TOK_PER_BLK;      // 512 blocks, exact fit -> EXEC all ones
    router_wmma_kernel<<<grid, 256, 0, stream>>>(x, Wg, outCombine, outIdx, outProbs);
}